// TaylorMap_52664888984164
// MI455X (gfx1250) — compile-verified
//
#include <hip/hip_runtime.h>

typedef __attribute__((ext_vector_type(2))) float v2f;
typedef __attribute__((ext_vector_type(8))) float v8f;

#define N_FEATURES  32
#define BATCH       8192
#define NPOLY       6545            // 1 + 32 + 528 + 5984
#define KCHUNK      128
#define NCHUNKS     52
#define KPAD        (KCHUNK * NCHUNKS)  // 6656
#define ROWS_PER_WG 64
#define XSTRIDE     37              // odd -> 32-lane row-strided reads conflict-free
#define ASTRIDE     (KCHUNK + 4)    // 132 = 4 (mod 64) -> conflict-free b64 reads
#define WTSTRIDE    (KCHUNK + 4)

// ---------------------------------------------------------------------------
// Kernel 1: decode poly index k -> packed (f,g,h) feature indices.
//   monomial(k) = x[f]*x[g]*x[h] with x[32]==1.0 (identity), x[33]==0.0 (K pad)
// ---------------------------------------------------------------------------
__global__ void build_mono_table(unsigned int* __restrict__ mono, int kpad) {
    int k = blockIdx.x * blockDim.x + threadIdx.x;
    if (k >= kpad) return;
    unsigned int f = 33u, g = 33u, h = 33u;     // default: zero (K padding)
    if (k == 0) {
        f = 32u; g = 32u; h = 32u;              // constant term 1
    } else if (k <= 32) {
        f = (unsigned)(k - 1); g = 32u; h = 32u;
    } else if (k < 561) {
        int q = k - 33;
        int a = 0;
        while ((a + 1) * (a + 2) / 2 <= q) ++a; // triangular inverse
        f = (unsigned)a;
        g = (unsigned)(q - a * (a + 1) / 2);
        h = 32u;
    } else if (k < NPOLY) {
        int c = k - 561;
        int ff = 0;
        while ((ff + 1) * (ff + 2) * (ff + 3) / 6 <= c) ++ff; // tetrahedral inverse
        int m = c - ff * (ff + 1) * (ff + 2) / 6;
        int a = 0;
        while ((a + 1) * (a + 2) / 2 <= m) ++a;
        int b = m - a * (a + 1) / 2;
        f = (unsigned)ff; g = (unsigned)a; h = (unsigned)b;
    }
    mono[k] = f | (g << 8) | (h << 16);
}

// ---------------------------------------------------------------------------
// Kernel 2: out = x + poly(x) @ W via V_WMMA_F32_16X16X4_F32.
// Per K-chunk:
//   (a) issue transposing async DMA for the W chunk (global_load_async_to_lds_b32:
//       per-lane global addr W[gk][col] -> per-lane LDS addr wt[col][k_local]),
//   (b) cooperatively build the 64x128 monomial A-tile (overlaps the DMA),
//   (c) s_wait_asynccnt 0 + barrier,
//   (d) steady state: ds_load_2addr_b64 pairs + v_wmma per K=4.
// ---------------------------------------------------------------------------
__global__ __launch_bounds__(256) void taylor_wmma_f32(
    const float* __restrict__ x, const float* __restrict__ W,
    const unsigned int* __restrict__ mono, float* __restrict__ out)
{
    __shared__ float xs[ROWS_PER_WG * XSTRIDE];   //  9472 B : x tile + sentinels
    __shared__ float ac[ROWS_PER_WG * ASTRIDE];   // 33792 B : monomial A chunk [row][k]
    __shared__ float wt[N_FEATURES * WTSTRIDE];   // 16896 B : W chunk transposed [col][k]

    const int tid  = threadIdx.x;
    const int lane = tid & 31;
    const int wave = tid >> 5;
    const int row0 = blockIdx.x * ROWS_PER_WG;

    // ---- stage x tile: 64 rows x 32 cols (512 float4, 2 per thread) ----
    #pragma unroll
    for (int t = 0; t < 2; ++t) {
        int i4 = tid + t * 256;                 // 0..511
        int r  = i4 >> 3;                       // 8 float4 per row
        int c4 = i4 & 7;
        float4 v = ((const float4*)(x + (size_t)(row0 + r) * N_FEATURES))[c4];
        float* dst = &xs[r * XSTRIDE + c4 * 4];
        dst[0] = v.x; dst[1] = v.y; dst[2] = v.z; dst[3] = v.w;
    }
    if (tid < ROWS_PER_WG) {                    // sentinel slots
        xs[tid * XSTRIDE + 32] = 1.0f;          // identity factor
        xs[tid * XSTRIDE + 33] = 0.0f;          // K-padding factor
    }

    // ---- per-wave fragment bases (steady-state WMMA loop) ----
    const int rowTile = wave >> 1;              // 0..3
    const int colHalf = wave & 1;               // 0..1
    const int l15     = lane & 15;
    const int hi      = lane >> 4;              // K base 0 or 2 (ISA A/B layout)
    const int kb      = hi * 2;
    const float* aBase = &ac[(rowTile * 16 + l15) * ASTRIDE + kb];
    const float* bBase = &wt[(colHalf * 16 + l15) * WTSTRIDE + kb];
    const int col      = colHalf * 16 + l15;    // output column 0..31

    // ---- per-thread build-phase bases: rows (lane) and (lane+32) ----
    const float* xrow0 = &xs[lane * XSTRIDE];
    const float* xrow1 = &xs[(lane + 32) * XSTRIDE];
    float* arow0 = &ac[lane * ASTRIDE + wave * 16];
    float* arow1 = &ac[(lane + 32) * ASTRIDE + wave * 16];

    // ---- async-DMA bases for the transposing W staging ----
    // lane -> W column; wave -> 16-row K slice. LDS dest: wt[col][wave*16 + t]
    const unsigned long long Wbase = (unsigned long long)(uintptr_t)W;
    const unsigned int wtLdsBase =                       // LDS byte offset (addr[31:0])
        (unsigned int)(uintptr_t)&wt[lane * WTSTRIDE + wave * 16];

    v8f acc = {0.f, 0.f, 0.f, 0.f, 0.f, 0.f, 0.f, 0.f};

    for (int kc = 0; kc < KPAD; kc += KCHUNK) {
        __syncthreads();                        // previous chunk fully consumed

        // ---- (a) transposing async DMA: W[kc + wave*16 + t][lane] -> wt ----
        {
            unsigned int gvoff =                // byte offset of W[kc+wave*16][lane]
                (unsigned int)(((kc + wave * 16) * N_FEATURES + lane) * 4);
            #pragma unroll
            for (int t = 0; t < 16; ++t) {
                int gk = kc + wave * 16 + t;    // uniform per wave: no divergence
                if (gk < NPOLY) {
                    unsigned int ldsa = wtLdsBase + (unsigned int)(t * 4);
                    unsigned int goff = gvoff + (unsigned int)(t * N_FEATURES * 4);
                    asm volatile(
                        "global_load_async_to_lds_b32 %0, %1, %2"
                        :: "v"(ldsa), "v"(goff), "s"(Wbase)
                        : "memory");
                } else {
                    wt[lane * WTSTRIDE + wave * 16 + t] = 0.f;  // K tail padding
                }
            }
        }

        // ---- prefetch next W chunk into cache (global_prefetch_b8) ----
        {
            int nk = kc + KCHUNK;
            if (nk < KPAD) {
                const float* p = W + (size_t)nk * N_FEATURES + (size_t)tid * 16;
                if (nk * N_FEATURES + tid * 16 < NPOLY * N_FEATURES)
                    __builtin_prefetch(p, 0, 3);
            }
        }

        // ---- (b) build monomial A-tile (overlaps the async DMA) ----
        #pragma unroll
        for (int i = 0; i < 16; ++i) {
            int ks = __builtin_amdgcn_readfirstlane(kc + wave * 16 + i);
            unsigned int m = mono[ks];          // wave-uniform -> scalar load
            int f = (int)(m & 0xffu);
            int g = (int)((m >> 8) & 0xffu);
            int h = (int)((m >> 16) & 0xffu);
            arow0[i] = xrow0[f] * xrow0[g] * xrow0[h];
            arow1[i] = xrow1[f] * xrow1[g] * xrow1[h];
        }

        // ---- (c) DMA completion + publish ----
        asm volatile("s_wait_asynccnt 0" ::: "memory");
        __syncthreads();

        // ---- (d) steady state: 32 WMMAs, ds_load_2addr_b64 fragment pairs ----
        #pragma unroll 8
        for (int k2 = 0; k2 < KCHUNK; k2 += 4) {
            v2f a = *(const v2f*)(aBase + k2);  // A frag: rows, K pair (b64)
            v2f b = *(const v2f*)(bBase + k2);  // B frag: cols, K pair (b64)
            acc = __builtin_amdgcn_wmma_f32_16x16x4_f32(
                false, a, false, b, (short)0, acc, false, false);
        }
    }

    // ---- epilogue: residual add + store. C layout: VGPR v -> M = v + hi*8 ----
    #pragma unroll
    for (int v = 0; v < 8; ++v) {
        int mr   = v + hi * 8;
        int grow = row0 + rowTile * 16 + mr;
        out[(size_t)grow * N_FEATURES + col] =
            acc[v] + xs[(rowTile * 16 + mr) * XSTRIDE + col];
    }
}

extern "C" void kernel_launch(void* const* d_in, const int* in_sizes, int n_in,
                              void* d_out, int out_size, void* d_ws, size_t ws_size,
                              hipStream_t stream) {
    const float* x = (const float*)d_in[0];   // [8192, 32]
    const float* W = (const float*)d_in[1];   // [6545, 32]
    float* out = (float*)d_out;               // [8192, 32]
    unsigned int* mono = (unsigned int*)d_ws; // KPAD u32 entries (26 KB)

    build_mono_table<<<(KPAD + 255) / 256, 256, 0, stream>>>(mono, KPAD);
    taylor_wmma_f32<<<BATCH / ROWS_PER_WG, 256, 0, stream>>>(x, W, mono, out);
}